// VectorQuantizer_38465727103636
// MI455X (gfx1250) — compile-verified
//
#include <hip/hip_runtime.h>

typedef __attribute__((ext_vector_type(2))) float v2f;
typedef __attribute__((ext_vector_type(4))) float v4f;
typedef __attribute__((ext_vector_type(8))) float v8f;

#define NTOK (16 * 4096)   // B*T = 65536 tokens
#define DIMS 64            // feature dim
#define NCB  1024          // codebook entries
#define MTOK 32            // tokens per wave (two 16-row WMMA tiles)
#define LDS_STRIDE 68      // floats per staged codebook row (bank-conflict-free)

// ---------------------------------------------------------------------------
// Prep: per-centroid squared norm into workspace + passthrough codebook copy
// ---------------------------------------------------------------------------
__global__ void __launch_bounds__(256)
vq_prep_kernel(const float* __restrict__ codebook,
               float* __restrict__ cnorm,
               float* __restrict__ out_cb) {
    int k = blockIdx.x * blockDim.x + threadIdx.x;
    if (k >= NCB) return;
    const v4f* src = reinterpret_cast<const v4f*>(codebook + (size_t)k * DIMS);
    v4f* dst = reinterpret_cast<v4f*>(out_cb + (size_t)k * DIMS);
    float s = 0.0f;
#pragma unroll
    for (int i = 0; i < DIMS / 4; ++i) {
        v4f v = src[i];
        dst[i] = v;
        s = fmaf(v.x, v.x, s);
        s = fmaf(v.y, v.y, s);
        s = fmaf(v.z, v.z, s);
        s = fmaf(v.w, v.w, s);
    }
    cnorm[k] = s;
}

// ---------------------------------------------------------------------------
// Async-copy one 16x64 codebook tile into an LDS buffer.
// 256 threads x 16B each = 4KB tile; one async instruction per thread.
// ---------------------------------------------------------------------------
__device__ __forceinline__ void async_stage_tile(const float* __restrict__ codebook,
                                                 int n0, float* lds_buf, int tid) {
    const int row = tid >> 4;        // 0..15 : codebook row within tile
    const int seg = tid & 15;        // 0..15 : 16B chunk within row
    unsigned long long ga =
        (unsigned long long)(uintptr_t)(codebook + (size_t)(n0 + row) * DIMS + seg * 4);
    unsigned lofs = (unsigned)(uintptr_t)(lds_buf + row * LDS_STRIDE + seg * 4);
    asm volatile("global_load_async_to_lds_b128 %0, %1, off"
                 :: "v"(lofs), "v"(ga) : "memory");
}

__device__ __forceinline__ void wait_async_done() {
    asm volatile("s_wait_asynccnt 0x0" ::: "memory");
}

// ---------------------------------------------------------------------------
// Main: fused fp32 WMMA distance matmul + row argmin + gather + loss.
// Block = 8 waves x 32 tokens = 256 tokens; all waves sweep the 64 codebook
// tiles in lockstep through a double-buffered LDS stage filled with
// global_load_async_to_lds_b128.
// ---------------------------------------------------------------------------
__global__ void __launch_bounds__(256)
vq_main_kernel(const float* __restrict__ inputs,
               const float* __restrict__ codebook,
               const float* __restrict__ cnorm,
               float* __restrict__ out_q,
               float* __restrict__ out_loss,
               float* __restrict__ out_idx) {
    __shared__ float smem[2][16 * LDS_STRIDE];

    const int tid  = threadIdx.x;
    const int lane = tid & 31;
    const int wave = tid >> 5;
    const int tile = blockIdx.x * 8 + wave;       // 0 .. 2047
    const int m0   = tile * MTOK;                 // first token of this tile

    const int half = lane >> 4;                   // 0: K pair (0,1), 1: K pair (2,3)
    const int l16  = lane & 15;

    // ---- A fragments: 2 x (16 tokens x 64 dims), 16 K-steps of 16x4 fp32 ----
    // 16x4 fp32 A layout: lanes 0-15 hold K=(0,1), lanes 16-31 hold K=(2,3),
    // row M = lane & 15 -> contiguous float2 per lane per step.
    v2f a0[16], a1[16];
    {
        const float* ar0 = inputs + (size_t)(m0 + l16) * DIMS + 2 * half;
        const float* ar1 = inputs + (size_t)(m0 + 16 + l16) * DIMS + 2 * half;
#pragma unroll
        for (int kk = 0; kk < 16; ++kk) {
            a0[kk] = *reinterpret_cast<const v2f*>(ar0 + kk * 4);
            a1[kk] = *reinterpret_cast<const v2f*>(ar1 + kk * 4);
        }
    }

    float bestV0[8], bestV1[8];
    int   bestI0[8], bestI1[8];
#pragma unroll
    for (int r = 0; r < 8; ++r) {
        bestV0[r] = 3.0e38f; bestI0[r] = 0;
        bestV1[r] = 3.0e38f; bestI1[r] = 0;
    }

    // ---- prologue: stage tile 0 into buffer 0 ----
    async_stage_tile(codebook, 0, &smem[0][0], tid);
    wait_async_done();
    __syncthreads();

    // ---- sweep 64 centroid tiles of 16, double-buffered through LDS ----
    for (int it = 0; it < NCB / 16; ++it) {
        const int p  = it & 1;
        const int n0 = it * 16;
        const int n  = n0 + l16;                  // centroid column for this lane

        // kick off async fill of the other buffer for the next tile
        if (it + 1 < NCB / 16)
            async_stage_tile(codebook, n0 + 16, &smem[p ^ 1][0], tid);

        // B 4x16 fp32 fragments from LDS: lanes 0-15 K=(0,1), lanes 16-31
        // K=(2,3), column N = lane & 15. Stride 68 -> conflict-free ds_load_b64.
        const float* bbase = &smem[p][l16 * LDS_STRIDE + 2 * half];

        v8f acc0 = {};                            // 16x16 fp32 D tiles
        v8f acc1 = {};
#pragma unroll
        for (int kk = 0; kk < 16; ++kk) {
            v2f b = *reinterpret_cast<const v2f*>(bbase + kk * 4);
            acc0 = __builtin_amdgcn_wmma_f32_16x16x4_f32(
                false, a0[kk], false, b, (short)0, acc0, false, false);
            acc1 = __builtin_amdgcn_wmma_f32_16x16x4_f32(
                false, a1[kk], false, b, (short)0, acc1, false, false);
        }

        const float cn = cnorm[n];
#pragma unroll
        for (int r = 0; r < 8; ++r) {
            // score = -2*dot + ||c||^2 ; row M = r + 8*half (+16 for chain 1)
            float s0 = fmaf(-2.0f, acc0[r], cn);
            float s1 = fmaf(-2.0f, acc1[r], cn);
            if (s0 < bestV0[r]) { bestV0[r] = s0; bestI0[r] = n; }
            if (s1 < bestV1[r]) { bestV1[r] = s1; bestI1[r] = n; }
        }

        wait_async_done();     // next tile landed in LDS (this wave's issues)
        __syncthreads();       // visible to all waves; readers of buf p done
    }

    // ---- cross-lane argmin within each 16-lane half (tie -> lower index) ----
#pragma unroll
    for (int r = 0; r < 8; ++r) {
        float v0 = bestV0[r]; int i0 = bestI0[r];
        float v1 = bestV1[r]; int i1 = bestI1[r];
#pragma unroll
        for (int mask = 8; mask >= 1; mask >>= 1) {
            float ov0 = __shfl_xor(v0, mask, 32);
            int   oi0 = __shfl_xor(i0, mask, 32);
            float ov1 = __shfl_xor(v1, mask, 32);
            int   oi1 = __shfl_xor(i1, mask, 32);
            if (ov0 < v0 || (ov0 == v0 && oi0 < i0)) { v0 = ov0; i0 = oi0; }
            if (ov1 < v1 || (ov1 == v1 && oi1 < i1)) { v1 = ov1; i1 = oi1; }
        }
        bestI0[r] = i0;   // lane 0: token m0+r      lane 16: token m0+8+r
        bestI1[r] = i1;   // lane 0: token m0+16+r   lane 16: token m0+24+r
    }

    // nn_idx output (stored as float, one per token)
#pragma unroll
    for (int r = 0; r < 8; ++r) {
        if (lane == 0) {
            out_idx[m0 + r]      = (float)bestI0[r];
            out_idx[m0 + 16 + r] = (float)bestI1[r];
        }
        if (lane == 16) {
            out_idx[m0 + 8 + r]  = (float)bestI0[r];
            out_idx[m0 + 24 + r] = (float)bestI1[r];
        }
    }

    // broadcast the 32 winning indices to every lane
    int idxTok[MTOK];
#pragma unroll
    for (int r = 0; r < 8; ++r) {
        idxTok[r]      = __shfl(bestI0[r], 0, 32);
        idxTok[8 + r]  = __shfl(bestI0[r], 16, 32);
        idxTok[16 + r] = __shfl(bestI1[r], 0, 32);
        idxTok[24 + r] = __shfl(bestI1[r], 16, 32);
    }

    // ---- gather winning centroids + quantization loss ----
    // each lane owns 2 of the 64 dims -> 8B vector loads/stores
    const int d0 = lane * 2;
#pragma unroll
    for (int t = 0; t < MTOK; ++t) {
        const int tok = m0 + t;
        const int ci  = idxTok[t];
        v2f q = *reinterpret_cast<const v2f*>(codebook + (size_t)ci * DIMS + d0);
        v2f x = *reinterpret_cast<const v2f*>(inputs + (size_t)tok * DIMS + d0);
        v2f ls;
        ls.x = (q.x - x.x) * (q.x - x.x);
        ls.y = (q.y - x.y) * (q.y - x.y);
        *reinterpret_cast<v2f*>(out_q + (size_t)tok * DIMS + d0)    = q;
        *reinterpret_cast<v2f*>(out_loss + (size_t)tok * DIMS + d0) = ls;
    }
}

// ---------------------------------------------------------------------------
extern "C" void kernel_launch(void* const* d_in, const int* in_sizes, int n_in,
                              void* d_out, int out_size, void* d_ws, size_t ws_size,
                              hipStream_t stream) {
    const float* inputs   = (const float*)d_in[0];   // [16,4096,64] f32
    const float* codebook = (const float*)d_in[1];   // [1024,64]    f32

    float* out      = (float*)d_out;
    float* out_q    = out;                                   // [B,T,D]
    float* out_loss = out + (size_t)NTOK * DIMS;             // [B,T,D]
    float* out_idx  = out + (size_t)2 * NTOK * DIMS;         // [B,T]
    float* out_cb   = out_idx + NTOK;                        // [K,D]

    float* cnorm = (float*)d_ws;                             // 1024 floats

    vq_prep_kernel<<<NCB / 256, 256, 0, stream>>>(codebook, cnorm, out_cb);

    // 2048 tiles of 32 tokens, 8 waves per 256-thread block -> 256 blocks
    vq_main_kernel<<<(NTOK / MTOK) / 8, 256, 0, stream>>>(
        inputs, codebook, cnorm, out_q, out_loss, out_idx);
}